// MuyGPs_layer_78726750536169
// MI455X (gfx1250) — compile-verified
//
#include <hip/hip_runtime.h>

typedef __attribute__((ext_vector_type(2))) float v2f;
typedef __attribute__((ext_vector_type(8))) float v8f;

#define BATCH 4096
#define FEAT  40
#define KNN   30
#define RR    10
#define EPS_  1e-3f
#define SQRT3_ 1.7320508075688772f

#define XS 48   // LDS row stride for X (floats), avoids bank conflicts
#define GS 33   // LDS row stride for G / Kin (floats)

// One workgroup (128 threads = 4 wave32) per batch element.
__global__ __launch_bounds__(128) void muygps_batch_kernel(
    const float* __restrict__ x,            // [40000, 40]
    const float* __restrict__ length_scale, // [1]
    const float* __restrict__ targets,      // [B, 30, 10]
    const int*   __restrict__ batch_indices,// [B]
    const int*   __restrict__ nn_indices,   // [B, 30]
    float* __restrict__ out_pred,           // [B, 10]
    float* __restrict__ ws_sigma,           // [B, 10] partial sigma contributions
    float* __restrict__ ws_var)             // [B]     unscaled variances
{
    __shared__ float Xs[32 * XS];     // rows 0..29 = neighbors, 30 = query, 31 = zero
    __shared__ float G[32 * GS];      // Gram, then Kin, then Cholesky factor
    __shared__ float diag[32];
    __shared__ float Kc[32];          // Matern cross covariances
    __shared__ float rhs[KNN * 12];   // 11 RHS columns (stride 12)

    const int b    = blockIdx.x;
    const int tid  = threadIdx.x;
    const int lane = tid & 31;
    const int wave = tid >> 5;

    const float ls = length_scale[0];
    const int   bi = batch_indices[b];

    // ---- gather X rows into LDS ----
    for (int t = tid; t < 32 * FEAT; t += 128) {
        int row = t / FEAT, col = t % FEAT;
        float v = 0.f;
        if (row < KNN)       v = x[nn_indices[b * KNN + row] * FEAT + col];
        else if (row == 30)  v = x[bi * FEAT + col];
        Xs[row * XS + col] = v;
    }
    __syncthreads();

    // ---- Gram matrix G = X * X^T (32x32, K=40) via V_WMMA_F32_16X16X4_F32 ----
    {
        const int tM   = (wave >> 1) << 4;       // 0 or 16
        const int tN   = (wave & 1) << 4;        // 0 or 16
        const int mRow = tM + (lane & 15);
        const int nCol = tN + (lane & 15);
        const int kOff = (lane >> 4) << 1;       // 0 (lanes 0-15) or 2 (lanes 16-31)
        v8f acc = {};
        for (int k0 = 0; k0 < FEAT; k0 += 4) {
            v2f a, bb;
            a.x  = Xs[mRow * XS + k0 + kOff];
            a.y  = Xs[mRow * XS + k0 + kOff + 1];
            bb.x = Xs[nCol * XS + k0 + kOff];
            bb.y = Xs[nCol * XS + k0 + kOff + 1];
            acc = __builtin_amdgcn_wmma_f32_16x16x4_f32(
                false, a, false, bb, (short)0, acc, false, false);
        }
        // C/D layout: lanes 0-15 -> M = tM+r, lanes 16-31 -> M = tM+8+r; N = tN + lane%16
        const int rBase = tM + ((lane >> 4) << 3);
        const int cIdx  = tN + (lane & 15);
        #pragma unroll
        for (int r = 0; r < 8; ++r)
            G[(rBase + r) * GS + cIdx] = acc[r];
    }
    __syncthreads();

    if (tid < 32) diag[tid] = G[tid * GS + tid];
    __syncthreads();

    // ---- Kcross (uses row 30 of Gram) ----
    if (tid < KNN) {
        float sq = fmaxf(diag[30] + diag[tid] - 2.f * G[30 * GS + tid], 0.f);
        float d  = sqrtf(sq);
        float z  = SQRT3_ * d / ls;
        Kc[tid]  = (1.f + z) * __expf(-z);
    }
    // ---- Kin = matern(pair_sq) + eps*I, in-place over G ----
    for (int t = tid; t < KNN * KNN; t += 128) {
        int i = t / KNN, j = t % KNN;
        float v;
        if (i == j) v = 1.f + EPS_;            // matern(0) = 1
        else {
            float sq = fmaxf(diag[i] + diag[j] - 2.f * G[i * GS + j], 0.f);
            float d  = sqrtf(sq);
            float z  = SQRT3_ * d / ls;
            v = (1.f + z) * __expf(-z);
        }
        G[i * GS + j] = v;
    }
    // ---- rhs columns 0..9 = targets, 11 = pad ----
    for (int t = tid; t < KNN * 12; t += 128) {
        int row = t / 12, col = t % 12;
        float v = 0.f;
        if (col < RR) v = targets[(b * KNN + row) * RR + col];
        rhs[t] = v;
    }
    __syncthreads();
    if (tid < KNN) rhs[tid * 12 + RR] = Kc[tid];   // column 10 = Kcross
    __syncthreads();

    // ---- Cholesky (symmetric full-square trailing updates) ----
    for (int k = 0; k < KNN; ++k) {
        if (tid == 0) G[k * GS + k] = sqrtf(G[k * GS + k]);
        __syncthreads();
        float dk = G[k * GS + k];
        if (tid > k && tid < KNN) G[tid * GS + k] /= dk;
        __syncthreads();
        int m = KNN - 1 - k;
        for (int t = tid; t < m * m; t += 128) {
            int i = k + 1 + t / m, j = k + 1 + t % m;
            G[i * GS + j] -= G[i * GS + k] * G[j * GS + k];
        }
        __syncthreads();
    }

    // ---- triangular solves: one RHS column per thread (no cross-thread deps) ----
    if (tid < RR + 1) {
        // forward: L y = rhs
        for (int i = 0; i < KNN; ++i) {
            float s = rhs[i * 12 + tid];
            for (int j = 0; j < i; ++j) s -= G[i * GS + j] * rhs[j * 12 + tid];
            rhs[i * 12 + tid] = s / G[i * GS + i];
        }
        // backward: L^T x = y
        for (int i = KNN - 1; i >= 0; --i) {
            float s = rhs[i * 12 + tid];
            for (int j = i + 1; j < KNN; ++j) s -= G[j * GS + i] * rhs[j * 12 + tid];
            rhs[i * 12 + tid] = s / G[i * GS + i];
        }
    }
    __syncthreads();

    // ---- outputs ----
    if (tid < RR) {
        float p = 0.f, sp = 0.f;
        for (int k = 0; k < KNN; ++k) {
            float fy = rhs[k * 12 + tid];
            p  += Kc[k] * fy;
            sp += targets[(b * KNN + k) * RR + tid] * fy;
        }
        out_pred[b * RR + tid] = p;
        ws_sigma[b * RR + tid] = sp;
    }
    if (tid == RR) {
        float v = 0.f;
        for (int k = 0; k < KNN; ++k) v += Kc[k] * rhs[k * 12 + RR];
        ws_var[b] = 1.f - v;
    }
}

// Deterministic fixed-order reduction of sigma_sq + variance write-out.
__global__ __launch_bounds__(256) void muygps_reduce_kernel(
    const float* __restrict__ ws_sigma,  // [B, 10]
    const float* __restrict__ ws_var,    // [B]
    float* __restrict__ out_var)         // [B, 10]
{
    __shared__ float part[16][RR];
    __shared__ float ssq[RR];
    const int tid = threadIdx.x;

    if (tid < 16 * RR) {
        int r = tid % RR;
        int c = tid / RR;                 // 16 chunks of 256 batches
        float s = 0.f;
        for (int bb = c * 256; bb < (c + 1) * 256; ++bb)
            s += ws_sigma[bb * RR + r];
        part[c][r] = s;
    }
    __syncthreads();
    if (tid < RR) {
        float s = 0.f;
        for (int c = 0; c < 16; ++c) s += part[c][tid];
        ssq[tid] = s / (float)(BATCH * KNN);
    }
    __syncthreads();
    for (int t = tid; t < BATCH * RR; t += 256)
        out_var[t] = ws_var[t / RR] * ssq[t % RR];
}

extern "C" void kernel_launch(void* const* d_in, const int* in_sizes, int n_in,
                              void* d_out, int out_size, void* d_ws, size_t ws_size,
                              hipStream_t stream) {
    const float* x    = (const float*)d_in[0];
    const float* ls   = (const float*)d_in[1];
    const float* tg   = (const float*)d_in[2];
    const int*   bidx = (const int*)d_in[3];
    const int*   nidx = (const int*)d_in[4];

    float* out      = (float*)d_out;
    float* pred     = out;                    // [B,10]
    float* var      = out + BATCH * RR;       // [B,10]
    float* ws_sigma = (float*)d_ws;           // [B,10]
    float* ws_var   = ws_sigma + BATCH * RR;  // [B]

    muygps_batch_kernel<<<BATCH, 128, 0, stream>>>(x, ls, tg, bidx, nidx,
                                                   pred, ws_sigma, ws_var);
    muygps_reduce_kernel<<<1, 256, 0, stream>>>(ws_sigma, ws_var, var);
}